// TokenGNN_54176717472207
// MI455X (gfx1250) — compile-verified
//
#include <hip/hip_runtime.h>
#include <hip/hip_bf16.h>

#define NN 50000
#define EE 600000
#define ET (EE + NN)   // edges incl. self loops
#define HH 128
#define DD 128
#define OUTC (DD + 2 * HH)  // 384
#define NEG_SLOPE 0.2f

typedef float v2f __attribute__((ext_vector_type(2)));
typedef float v8f __attribute__((ext_vector_type(8)));

__device__ __forceinline__ float wred_sum(float v) {
  for (int o = 16; o > 0; o >>= 1) v += __shfl_xor(v, o, 32);
  return v;
}
__device__ __forceinline__ float wred_max(float v) {
  for (int o = 16; o > 0; o >>= 1) v = fmaxf(v, __shfl_xor(v, o, 32));
  return v;
}

// ---------------- graph preprocessing (built once, reused by both layers) ---

__global__ void tg_sum_ea(const float* __restrict__ ea, float* scal) {
  __shared__ float sh[256];
  float s = 0.f;
  for (int i = blockIdx.x * 256 + threadIdx.x; i < EE; i += 256 * gridDim.x)
    s += ea[i];
  sh[threadIdx.x] = s;
  __syncthreads();
  for (int o = 128; o > 0; o >>= 1) {
    if ((int)threadIdx.x < o) sh[threadIdx.x] += sh[threadIdx.x + o];
    __syncthreads();
  }
  if (threadIdx.x == 0) atomicAdd(&scal[0], sh[0]);
}

// scal[1] = mean(ea); scal[2] = dot(We0,ae0); scal[3] = dot(We1,ae1)
__global__ void tg_dots(const float* __restrict__ We0, const float* __restrict__ ae0,
                        const float* __restrict__ We1, const float* __restrict__ ae1,
                        float* scal) {
  __shared__ float sh[128];
  int t = threadIdx.x;
  sh[t] = We0[t] * ae0[t];
  __syncthreads();
  for (int o = 64; o > 0; o >>= 1) {
    if (t < o) sh[t] += sh[t + o];
    __syncthreads();
  }
  if (t == 0) scal[2] = sh[0];
  __syncthreads();
  sh[t] = We1[t] * ae1[t];
  __syncthreads();
  for (int o = 64; o > 0; o >>= 1) {
    if (t < o) sh[t] += sh[t + o];
    __syncthreads();
  }
  if (t == 0) {
    scal[3] = sh[0];
    scal[1] = scal[0] / (float)EE;  // mean edge attr (self-loop fill value)
  }
}

__global__ void tg_deg_init(int* deg) {
  int v = blockIdx.x * 256 + threadIdx.x;
  if (v < NN) deg[v] = 1;  // self loop
}

__global__ void tg_deg_count(const int* __restrict__ dst, int* deg) {
  int e = blockIdx.x * 256 + threadIdx.x;
  if (e < EE) atomicAdd(&deg[dst[e]], 1);
}

__global__ void tg_scan(const int* __restrict__ deg, int* __restrict__ rowptr, int n) {
  __shared__ int sh[1024];
  __shared__ int carry_s;
  int tid = threadIdx.x;
  if (tid == 0) carry_s = 0;
  __syncthreads();
  for (int base = 0; base < n; base += 1024) {
    int i = base + tid;
    int val = (i < n) ? deg[i] : 0;
    sh[tid] = val;
    __syncthreads();
    for (int off = 1; off < 1024; off <<= 1) {
      int t = (tid >= off) ? sh[tid - off] : 0;
      __syncthreads();
      sh[tid] += t;
      __syncthreads();
    }
    int incl = sh[tid];
    int carry = carry_s;
    if (i < n) rowptr[i] = carry + incl - val;  // exclusive scan
    int total = sh[1023];
    __syncthreads();
    if (tid == 0) carry_s = carry + total;
    __syncthreads();
  }
  if (tid == 0) rowptr[n] = carry_s;
}

__global__ void tg_cursor_copy(const int* __restrict__ rowptr, int* cursor) {
  int v = blockIdx.x * 256 + threadIdx.x;
  if (v < NN) cursor[v] = rowptr[v];
}

__global__ void tg_fill_edges(const int* __restrict__ src, const int* __restrict__ dst,
                              const float* __restrict__ ea, int* cursor,
                              int* __restrict__ col, float* __restrict__ eval) {
  int e = blockIdx.x * 256 + threadIdx.x;
  if (e >= EE) return;
  int d = dst[e];
  int pos = atomicAdd(&cursor[d], 1);
  col[pos] = src[e];
  eval[pos] = ea[e];
}

__global__ void tg_fill_self(int* cursor, int* __restrict__ col,
                             float* __restrict__ eval, const float* scal) {
  int v = blockIdx.x * 256 + threadIdx.x;
  if (v >= NN) return;
  int pos = atomicAdd(&cursor[v], 1);
  col[pos] = v;
  eval[pos] = scal[1];  // mean edge attr
}

__global__ void tg_copy_x(const float* __restrict__ x, float* __restrict__ outp) {
  int i = blockIdx.x * 256 + threadIdx.x;
  if (i >= NN * DD) return;
  int v = i >> 7, c = i & 127;
  outp[(size_t)v * OUTC + c] = x[i];
}

// ---------------- per-layer kernels ----------------------------------------

// hp[m,k] = sum_c A[m,c] * W[k,c]; A:[NN,128] row-major, W:[128,128] row-major.
// One wave per 16x16 output tile, fp32 WMMA (16x16x4). NN = 3125*16 exactly.
__global__ void tg_gemm(const float* __restrict__ A, const float* __restrict__ W,
                        float* __restrict__ hp) {
  int lane = threadIdx.x & 31;
  int wv = threadIdx.x >> 5;  // 0..7 -> column tile
  int m0 = blockIdx.x * 16;
  int k0 = wv * 16;
  int lr = lane & 15, lh = lane >> 4;
  // A tile lane layout (16x4 f32): lanes 0-15 rows m0+lr K={c0,c0+1},
  // lanes 16-31 rows m0+lr K={c0+2,c0+3}.  B (=W^T) mirrors with rows of W.
  const float* arow = A + (size_t)(m0 + lr) * 128 + lh * 2;
  const float* brow = W + (size_t)(k0 + lr) * 128 + lh * 2;
  v8f c = {};
#pragma unroll 4
  for (int c0 = 0; c0 < 128; c0 += 4) {
    v2f a = *(const v2f*)(arow + c0);
    v2f b = *(const v2f*)(brow + c0);
    c = __builtin_amdgcn_wmma_f32_16x16x4_f32(false, a, false, b, (short)0, c,
                                              false, false);
  }
  // C/D layout: VGPR r -> row (r + 8*lh), col lr within tile
  float* o = hp + (size_t)m0 * 128 + k0 + lr;
#pragma unroll
  for (int r = 0; r < 8; ++r) o[(size_t)(r + 8 * lh) * 128] = c[r];
}

// per-node attention logits: s_src[v] = hp[v]·a_s ; s_dst[v] = hp[v]·a_d
__global__ void tg_nodedot(const float* __restrict__ hp, const float* __restrict__ as,
                           const float* __restrict__ ad, float* __restrict__ ssrc,
                           float* __restrict__ sdst) {
  int lane = threadIdx.x & 31, wid = threadIdx.x >> 5;
  int v = blockIdx.x * 8 + wid;
  if (v >= NN) return;
  const float4 h4 = *(const float4*)(hp + (size_t)v * 128 + lane * 4);
  const float4 a4 = *(const float4*)(as + lane * 4);
  const float4 d4 = *(const float4*)(ad + lane * 4);
  float s1 = h4.x * a4.x + h4.y * a4.y + h4.z * a4.z + h4.w * a4.w;
  float s2 = h4.x * d4.x + h4.y * d4.y + h4.z * d4.z + h4.w * d4.w;
  s1 = wred_sum(s1);
  s2 = wred_sum(s2);
  if (lane == 0) {
    ssrc[v] = s1;
    sdst[v] = s2;
  }
}

// One wave per destination node: segment softmax + weighted gather-sum,
// bias + ReLU, write to next-layer input and to the JK-concat output.
__global__ void tg_attn(const float* __restrict__ hp, const int* __restrict__ rowptr,
                        const int* __restrict__ col, const float* __restrict__ eval,
                        float* alpha, const float* __restrict__ ssrc,
                        const float* __restrict__ sdst, const float* __restrict__ scal,
                        int ce_idx, const float* __restrict__ bias,
                        float* __restrict__ hnext, float* __restrict__ outp) {
  int lane = threadIdx.x & 31, wid = threadIdx.x >> 5;
  int v = blockIdx.x * 8 + wid;
  if (v >= NN) return;
  float ce = scal[ce_idx];
  int s = rowptr[v], e = rowptr[v + 1];
  float sdv = sdst[v];

  // pass 1: logits + leaky relu + running max
  float mx = -3.402823466e38f;
  for (int i = s + lane; i < e; i += 32) {
    float a = ssrc[col[i]] + sdv + ce * eval[i];
    a = a > 0.f ? a : NEG_SLOPE * a;
    alpha[i] = a;
    mx = fmaxf(mx, a);
  }
  mx = wred_max(mx);  // always finite: every node has a self loop

  // pass 2: exp + sum
  float sum = 0.f;
  for (int i = s + lane; i < e; i += 32) {
    float w = __expf(alpha[i] - mx);
    alpha[i] = w;
    sum += w;
  }
  sum = wred_sum(sum);
  float inv = 1.f / (sum + 1e-16f);

  // cross-lane store->load on alpha: drain VMEM stores (CDNA5 split counter)
  asm volatile("s_wait_storecnt 0" ::: "memory");

  // pass 3: weighted gather of hp rows (coalesced 512B per edge)
  float4 acc = make_float4(0.f, 0.f, 0.f, 0.f);
  for (int i = s; i < e; ++i) {
    float w = alpha[i] * inv;
    const float4 h4 = *(const float4*)(hp + (size_t)col[i] * 128 + lane * 4);
    acc.x += w * h4.x;
    acc.y += w * h4.y;
    acc.z += w * h4.z;
    acc.w += w * h4.w;
  }
  const float4 b4 = *(const float4*)(bias + lane * 4);
  acc.x = fmaxf(acc.x + b4.x, 0.f);
  acc.y = fmaxf(acc.y + b4.y, 0.f);
  acc.z = fmaxf(acc.z + b4.z, 0.f);
  acc.w = fmaxf(acc.w + b4.w, 0.f);
  *(float4*)(hnext + (size_t)v * 128 + lane * 4) = acc;
  *(float4*)(outp + (size_t)v * OUTC + lane * 4) = acc;
}

// ---------------------------------------------------------------------------

extern "C" void kernel_launch(void* const* d_in, const int* in_sizes, int n_in,
                              void* d_out, int out_size, void* d_ws, size_t ws_size,
                              hipStream_t stream) {
  const float* x   = (const float*)d_in[0];
  const int* eidx  = (const int*)d_in[1];
  const int* src   = eidx;
  const int* dst   = eidx + EE;
  const float* ea  = (const float*)d_in[2];
  const float* W0  = (const float*)d_in[3];
  const float* as0 = (const float*)d_in[4];
  const float* ad0 = (const float*)d_in[5];
  const float* We0 = (const float*)d_in[6];
  const float* ae0 = (const float*)d_in[7];
  const float* b0  = (const float*)d_in[8];
  const float* W1  = (const float*)d_in[9];
  const float* as1 = (const float*)d_in[10];
  const float* ad1 = (const float*)d_in[11];
  const float* We1 = (const float*)d_in[12];
  const float* ae1 = (const float*)d_in[13];
  const float* b1  = (const float*)d_in[14];
  float* out = (float*)d_out;

  // workspace carve-up (~60 MB)
  char* w = (char*)d_ws;
  float* hp    = (float*)w; w += (size_t)NN * HH * 4;
  float* hnext = (float*)w; w += (size_t)NN * HH * 4;
  float* ssrc  = (float*)w; w += (size_t)NN * 4;
  float* sdst  = (float*)w; w += (size_t)NN * 4;
  float* alpha = (float*)w; w += (size_t)ET * 4;
  float* eval  = (float*)w; w += (size_t)ET * 4;
  int*   col   = (int*)w;   w += (size_t)ET * 4;
  int*   deg   = (int*)w;   w += (size_t)NN * 4;
  int*   rowp  = (int*)w;   w += (size_t)(NN + 1) * 4;
  int*   curs  = (int*)w;   w += (size_t)NN * 4;
  float* scal  = (float*)w; w += 16 * 4;

  const int TB = 256;
  const int nodeB = (NN + TB - 1) / TB;          // 196
  const int edgeB = (EE + TB - 1) / TB;          // 2344
  const int waveNodeB = (NN + 7) / 8;            // 6250 (wave per node)

  // ---- graph build (every call; no state carried across calls) ----
  hipMemsetAsync(scal, 0, 16 * sizeof(float), stream);
  tg_sum_ea<<<256, 256, 0, stream>>>(ea, scal);
  tg_dots<<<1, 128, 0, stream>>>(We0, ae0, We1, ae1, scal);
  tg_deg_init<<<nodeB, TB, 0, stream>>>(deg);
  tg_deg_count<<<edgeB, TB, 0, stream>>>(dst, deg);
  tg_scan<<<1, 1024, 0, stream>>>(deg, rowp, NN);
  tg_cursor_copy<<<nodeB, TB, 0, stream>>>(rowp, curs);
  tg_fill_edges<<<edgeB, TB, 0, stream>>>(src, dst, ea, curs, col, eval);
  tg_fill_self<<<nodeB, TB, 0, stream>>>(curs, col, eval, scal);
  tg_copy_x<<<(NN * DD + TB - 1) / TB, TB, 0, stream>>>(x, out);

  // ---- layer 0 ----
  tg_gemm<<<NN / 16, 256, 0, stream>>>(x, W0, hp);
  tg_nodedot<<<waveNodeB, 256, 0, stream>>>(hp, as0, ad0, ssrc, sdst);
  tg_attn<<<waveNodeB, 256, 0, stream>>>(hp, rowp, col, eval, alpha, ssrc, sdst,
                                         scal, 2, b0, hnext, out + DD);

  // ---- layer 1 ----
  tg_gemm<<<NN / 16, 256, 0, stream>>>(hnext, W1, hp);
  tg_nodedot<<<waveNodeB, 256, 0, stream>>>(hp, as1, ad1, ssrc, sdst);
  tg_attn<<<waveNodeB, 256, 0, stream>>>(hp, rowp, col, eval, alpha, ssrc, sdst,
                                         scal, 3, b1, hnext, out + DD + HH);
}